// MMD_88570815578382
// MI455X (gfx1250) — compile-verified
//
#include <hip/hip_runtime.h>

typedef __attribute__((ext_vector_type(2))) float v2f;
typedef __attribute__((ext_vector_type(8))) float v8f;

#define D_DIM 16
#define TILE 16
#define JCHUNK 64
#define WAVES_PER_BLOCK 8

// -log2(e) / (2*bw^2) for bw = 1e-2, 1e-1, 1e0, 1e1
#define E_BW0 (-7213.4754f)      // -1.44269504f * 5000.f
#define E_BW1 (-72.134754f)      // -1.44269504f * 50.f
#define E_BW2 (-0.72134752f)     // -1.44269504f * 0.5f
#define E_BW3 (-0.0072134752f)   // -1.44269504f * 0.005f

__device__ __forceinline__ const float* row_ptr(const float* __restrict__ X,
                                                const float* __restrict__ Y,
                                                int nx, int row) {
  return (row < nx) ? (X + (size_t)row * D_DIM)
                    : (Y + (size_t)(row - nx) * D_DIM);
}

// Kernel 1: row squared-norms into workspace; zero the output scalar.
__global__ void mmd_sq_kernel(const float* __restrict__ X, const float* __restrict__ Y,
                              float* __restrict__ sq, float* __restrict__ out,
                              int nx, int N) {
  int i = blockIdx.x * blockDim.x + threadIdx.x;
  if (i == 0) out[0] = 0.0f;
  if (i < N) {
    const float* p = row_ptr(X, Y, nx, i);
    float s = 0.f;
#pragma unroll
    for (int k = 0; k < D_DIM; ++k) s = fmaf(p[k], p[k], s);
    sq[i] = s;
  }
}

// Kernel 2: one 16x16 Gram tile per wave per j-iteration via chained
// V_WMMA_F32_16X16X4_F32, fused RBF-sum epilogue, weighted reduction.
__global__ __launch_bounds__(256) void mmd_tile_kernel(
    const float* __restrict__ X, const float* __restrict__ Y,
    const float* __restrict__ sq, float* __restrict__ out,
    int nx, int ny, int N) {
  const int tilesN = N / TILE;
  const int numJ   = (tilesN + JCHUNK - 1) / JCHUNK;
  const int wave   = blockIdx.x * WAVES_PER_BLOCK + (int)(threadIdx.x >> 5);
  const int lane   = threadIdx.x & 31;
  const int half   = lane >> 4;   // 0: lanes 0-15, 1: lanes 16-31
  const int l15    = lane & 15;

  float acc = 0.0f;
  const int ti = wave / numJ;
  const int jc = wave % numJ;

  if (ti < tilesN) {  // wave-uniform branch: EXEC all-1s inside
    const float winx =  1.0f / (float)nx;
    const float winy = -1.0f / (float)ny;

    // A tile: row (ti*16 + l15); lanes 0-15 carry K={4kk,4kk+1}, 16-31 K={4kk+2,4kk+3}
    const float* Arow = row_ptr(X, Y, nx, ti * TILE + l15);
    const v2f a0 = *(const v2f*)(Arow + 0 * 4 + half * 2);
    const v2f a1 = *(const v2f*)(Arow + 1 * 4 + half * 2);
    const v2f a2 = *(const v2f*)(Arow + 2 * 4 + half * 2);
    const v2f a3 = *(const v2f*)(Arow + 3 * 4 + half * 2);

    // Row norms + MMD weights for the 8 output rows this lane owns
    float sqr[8], wr[8];
#pragma unroll
    for (int v = 0; v < 8; ++v) {
      int r = ti * TILE + v + 8 * half;   // D-matrix VGPR-v row mapping
      sqr[v] = sq[r];
      wr[v]  = (r < nx) ? winx : winy;
    }

    const int j0 = jc * JCHUNK;
    const int j1 = (j0 + JCHUNK < tilesN) ? (j0 + JCHUNK) : tilesN;
    for (int tj = j0; tj < j1; ++tj) {
      const int colB = tj * TILE + l15;
      const float* Brow = row_ptr(X, Y, nx, colB);
      const v2f b0 = *(const v2f*)(Brow + 0 * 4 + half * 2);
      const v2f b1 = *(const v2f*)(Brow + 1 * 4 + half * 2);
      const v2f b2 = *(const v2f*)(Brow + 2 * 4 + half * 2);
      const v2f b3 = *(const v2f*)(Brow + 3 * 4 + half * 2);

      v8f c = {0.f, 0.f, 0.f, 0.f, 0.f, 0.f, 0.f, 0.f};
      c = __builtin_amdgcn_wmma_f32_16x16x4_f32(false, a0, false, b0, (short)0, c, false, false);
      c = __builtin_amdgcn_wmma_f32_16x16x4_f32(false, a1, false, b1, (short)0, c, false, false);
      c = __builtin_amdgcn_wmma_f32_16x16x4_f32(false, a2, false, b2, (short)0, c, false, false);
      c = __builtin_amdgcn_wmma_f32_16x16x4_f32(false, a3, false, b3, (short)0, c, false, false);

      const float sqc = sq[colB];
      const float wc  = (colB < nx) ? winx : winy;

      float tileAcc = 0.0f;
#pragma unroll
      for (int v = 0; v < 8; ++v) {
        float t = fmaf(-2.0f, c[v], sqr[v] + sqc);  // squared distance
        t = fmaxf(t, 0.0f);
        float ks = __builtin_amdgcn_exp2f(t * E_BW0)
                 + __builtin_amdgcn_exp2f(t * E_BW1)
                 + __builtin_amdgcn_exp2f(t * E_BW2)
                 + __builtin_amdgcn_exp2f(t * E_BW3);
        tileAcc = fmaf(wr[v], ks, tileAcc);
      }
      acc = fmaf(wc, tileAcc, acc);
    }
  }

  // Block reduction: LDS atomic per thread, one global atomic per block.
  __shared__ float red;
  if (threadIdx.x == 0) red = 0.0f;
  __syncthreads();
  atomicAdd(&red, acc);
  __syncthreads();
  if (threadIdx.x == 0) atomicAdd(out, red);
}

extern "C" void kernel_launch(void* const* d_in, const int* in_sizes, int n_in,
                              void* d_out, int out_size, void* d_ws, size_t ws_size,
                              hipStream_t stream) {
  const float* X = (const float*)d_in[0];
  const float* Y = (const float*)d_in[1];
  const int nx = in_sizes[0] / D_DIM;   // 6144
  const int ny = in_sizes[1] / D_DIM;   // 6144
  const int N  = nx + ny;               // 12288

  float* sq  = (float*)d_ws;            // N floats of scratch
  float* out = (float*)d_out;

  mmd_sq_kernel<<<(N + 255) / 256, 256, 0, stream>>>(X, Y, sq, out, nx, N);

  const int tilesN     = N / TILE;                          // 768
  const int numJ       = (tilesN + JCHUNK - 1) / JCHUNK;    // 12
  const int totalWaves = tilesN * numJ;                     // 9216
  const int blocks     = (totalWaves + WAVES_PER_BLOCK - 1) / WAVES_PER_BLOCK;  // 1152

  mmd_tile_kernel<<<blocks, 256, 0, stream>>>(X, Y, sq, out, nx, ny, N);
}